// TripletLoss_28991029248550
// MI455X (gfx1250) — compile-verified
//
#include <hip/hip_runtime.h>

typedef __attribute__((ext_vector_type(2))) float v2f;
typedef __attribute__((ext_vector_type(8))) float v8f;

#define N 8192
#define D 128
#define NT (N / 16)          // 512 tiles per dimension
#define S 4                  // column splits (parallelism across the col range)
#define WPB 8                // waves per block (256 threads)
#define MARGIN 0.3f
#define LOSS_FLOOR 1e-6f

// ---------------------------------------------------------------- sq norms
__global__ void __launch_bounds__(256) tl_sqnorm(const float* __restrict__ x,
                                                 float* __restrict__ sq) {
    int row = blockIdx.x * blockDim.x + threadIdx.x;
    if (row >= N) return;
    const float4* p = (const float4*)(x + (size_t)row * D);
    float acc = 0.f;
#pragma unroll
    for (int i = 0; i < D / 4; ++i) {
        float4 v = p[i];
        acc += v.x * v.x + v.y * v.y + v.z * v.z + v.w * v.w;
    }
    sq[row] = acc;
}

__global__ void tl_init_acc(float* acc) { acc[0] = 0.f; acc[1] = 0.f; }

// --------------------------------------------- fused Gram + hard mining
// One wave owns one 16-row tile and one of S column ranges.
// Gram tile via v_wmma_f32_16x16x4_f32 (exact f32). Two independent
// accumulator chains (even/odd K steps) break the 32-deep WMMA->WMMA RAW
// dependency through the XDL pipe; distances + masked max/min are folded
// immediately so the 8192x8192 matrix is never materialized.
__global__ void __launch_bounds__(256) tl_hardmine(
    const float* __restrict__ x, const int* __restrict__ labels,
    const float* __restrict__ sq, float* __restrict__ hp_out,
    float* __restrict__ hn_out) {
    const float INF = __builtin_inff();
    const int lane  = threadIdx.x & 31;
    const int wave  = threadIdx.x >> 5;
    const int wg    = blockIdx.x * WPB + wave;   // global wave id
    const int rt    = wg % NT;                   // row tile
    const int cs    = wg / NT;                   // column split
    const int rowBase = rt * 16;
    const int nlane = lane & 15;
    const int hi    = lane >> 4;

    // ---- A tile preload: lane holds row (rowBase + nlane), K pairs at hi*2
    const int aRow = rowBase + nlane;
    const float* arow = x + (size_t)aRow * D + hi * 2;
    v2f areg[32];
#pragma unroll
    for (int k = 0; k < 32; ++k) areg[k] = *(const v2f*)(arow + k * 4);

    // ---- per-lane row metadata for the 8 C/D rows this lane contributes to
    float sqr[8];
    int   lblr[8];
#pragma unroll
    for (int v = 0; v < 8; ++v) {
        int r   = rowBase + v + hi * 8;
        sqr[v]  = sq[r];
        lblr[v] = labels[r];
    }

    float hp[8], hn[8];
#pragma unroll
    for (int v = 0; v < 8; ++v) { hp[v] = -INF; hn[v] = INF; }

    const int ct0 = cs * (NT / S);
    const int ct1 = ct0 + (NT / S);
    for (int ct = ct0; ct < ct1; ++ct) {
        const int colBase = ct * 16;
        const int bCol    = colBase + nlane;     // B lane column
        const float* brow = x + (size_t)bCol * D + hi * 2;

        v8f c0 = {};
        v8f c1 = {};
#pragma unroll
        for (int k = 0; k < 32; k += 2) {
            v2f b0 = *(const v2f*)(brow + k * 4);
            v2f b1 = *(const v2f*)(brow + k * 4 + 4);
            c0 = __builtin_amdgcn_wmma_f32_16x16x4_f32(
                false, areg[k], false, b0, (short)0, c0, false, false);
            c1 = __builtin_amdgcn_wmma_f32_16x16x4_f32(
                false, areg[k + 1], false, b1, (short)0, c1, false, false);
        }
        v8f c = c0 + c1;

        const float sqc  = sq[bCol];
        const int   lblc = labels[bCol];
#pragma unroll
        for (int v = 0; v < 8; ++v) {
            float d2   = sqr[v] + sqc - 2.0f * c[v];
            float dist = __builtin_sqrtf(fmaxf(d2, 1e-12f));
            int   row  = rowBase + v + hi * 8;
            bool  same = (lblr[v] == lblc);
            if (same) {
                if (row != bCol) hp[v] = fmaxf(hp[v], dist);
            } else {
                hn[v] = fminf(hn[v], dist);
            }
        }
    }

    // ---- reduce over the 16 lanes sharing each row (xor tree in half-wave)
#pragma unroll
    for (int v = 0; v < 8; ++v) {
#pragma unroll
        for (int off = 1; off < 16; off <<= 1) {
            hp[v] = fmaxf(hp[v], __shfl_xor(hp[v], off, 32));
            hn[v] = fminf(hn[v], __shfl_xor(hn[v], off, 32));
        }
    }
    if (nlane == 0) {
#pragma unroll
        for (int v = 0; v < 8; ++v) {
            int row = rowBase + v + hi * 8;
            hp_out[(size_t)cs * N + row] = hp[v];
            hn_out[(size_t)cs * N + row] = hn[v];
        }
    }
}

// ---------------------------------------------------------------- reduce
__global__ void __launch_bounds__(256) tl_reduce(const float* __restrict__ hp,
                                                 const float* __restrict__ hn,
                                                 float* __restrict__ acc) {
    const float INF = __builtin_inff();
    int row = blockIdx.x * blockDim.x + threadIdx.x;
    float per = 0.f, cnt = 0.f;
    if (row < N) {
        float p = -INF, q = INF;
#pragma unroll
        for (int s = 0; s < S; ++s) {
            p = fmaxf(p, hp[(size_t)s * N + row]);
            q = fminf(q, hn[(size_t)s * N + row]);
        }
        if (p != -INF && q != INF) {
            per = fmaxf(p - q + MARGIN, LOSS_FLOOR);
            cnt = 1.f;
        }
    }
#pragma unroll
    for (int off = 16; off > 0; off >>= 1) {
        per += __shfl_down(per, off, 32);
        cnt += __shfl_down(cnt, off, 32);
    }
    if ((threadIdx.x & 31) == 0) {
        atomicAdd(&acc[0], per);
        atomicAdd(&acc[1], cnt);
    }
}

__global__ void tl_finalize(const float* __restrict__ acc,
                            float* __restrict__ out) {
    out[0] = (acc[1] > 0.f) ? (acc[0] / acc[1]) : 0.f;
}

// ---------------------------------------------------------------- launch
extern "C" void kernel_launch(void* const* d_in, const int* in_sizes, int n_in,
                              void* d_out, int out_size, void* d_ws,
                              size_t ws_size, hipStream_t stream) {
    (void)in_sizes; (void)n_in; (void)out_size; (void)ws_size;
    const float* x      = (const float*)d_in[0];
    const int*   labels = (const int*)d_in[1];
    float*       out    = (float*)d_out;

    char*  ws  = (char*)d_ws;
    float* sq  = (float*)ws;                                   // N floats
    float* acc = (float*)(ws + (size_t)N * sizeof(float));     // 2 floats
    float* hp  = (float*)(ws + (size_t)N * sizeof(float) + 256);
    float* hn  = hp + (size_t)S * N;

    tl_sqnorm<<<N / 256, 256, 0, stream>>>(x, sq);
    tl_init_acc<<<1, 1, 0, stream>>>(acc);
    tl_hardmine<<<(NT * S) / WPB, 256, 0, stream>>>(x, labels, sq, hp, hn);
    tl_reduce<<<N / 256, 256, 0, stream>>>(hp, hn, acc);
    tl_finalize<<<1, 1, 0, stream>>>(acc, out);
}